// RankDetrTransformer_69458211111580
// MI455X (gfx1250) — compile-verified
//
#include <hip/hip_runtime.h>
#include <stdint.h>

// ---- Problem constants (static in the reference's setup_inputs) -------------
#define NQ   300               // Len_q == output trailing dim
#define NH   8                 // heads
#define NLV  4                 // levels
#define NP   4                 // points
#define S_TOT 22223            // sum of level H*W
#define ROWS_PER_BLK 48        // rows of M reduced into one output column
#define SAMPLES_PER_BLK (ROWS_PER_BLK * NLV * NP)   // 768

// ---- Transpose tiling -------------------------------------------------------
#define TS     32              // s-values per transpose tile
#define TROWS  NQ              // 300 b-rows per tile
#define TPITCH 33              // padded LDS row pitch in DWORDs (odd -> no bank conflicts)

typedef uint32_t u32x4 __attribute__((ext_vector_type(4)));
typedef int32_t  i32x8 __attribute__((ext_vector_type(8)));
typedef int32_t  i32x4 __attribute__((ext_vector_type(4)));

#if defined(__has_builtin)
#if __has_builtin(__builtin_amdgcn_tensor_load_to_lds) && \
    __has_builtin(__builtin_amdgcn_tensor_store_from_lds) && \
    __has_builtin(__builtin_amdgcn_s_wait_tensorcnt)
#define HAVE_TDM 1
#endif
#endif
#ifndef HAVE_TDM
#define HAVE_TDM 0
#endif

#if HAVE_TDM
// D# group 0: count=1 (valid, user mode), lds_addr, 57-bit global_addr, type=2.
__device__ __forceinline__ u32x4 tdm_group0(uint32_t lds_off, uint64_t gaddr) {
    u32x4 g;
    g.x = 0x1u;                                                  // count=1
    g.y = lds_off;                                               // lds_addr[31:0]
    g.z = (uint32_t)gaddr;                                       // global_addr[31:0]
    g.w = (uint32_t)((gaddr >> 32) & 0x01FFFFFFu) | (2u << 30);  // global_addr[56:32] | type=2
    return g;
}
#endif

// =============================================================================
// Kernel 1: bilinear scatter into an LDS-resident column, then bulk write-out.
// Column b reduces M-rows m in [48*b, 48*b+48); m = (l*NH + h)*NQ + q.
//   contiguous==1 : write column to dst + b*S_TOT (workspace, coalesced / TDM)
//   contiguous==0 : write column strided to dst + s*NQ + b (direct final layout)
// =============================================================================
__global__ void msda_scatter_kernel(const float* __restrict__ sloc,
                                    const float* __restrict__ awgt,
                                    float* __restrict__ dst,
                                    int contiguous) {
    extern __shared__ float smem[];   // S_TOT floats (88.9 KB of the WGP's 320 KB)
    const int b    = blockIdx.x;
    const int tid  = threadIdx.x;
    const int nthr = blockDim.x;

    // ---- Phase 1: zero the LDS column --------------------------------------
    for (int s = tid; s < S_TOT; s += nthr) smem[s] = 0.0f;
    __syncthreads();

    // FPN levels for 800x1333 image at strides 8/16/32/64
    const int Wl [NLV] = {167, 84, 42, 21};
    const int Hl [NLV] = {100, 50, 25, 13};
    const int LSI[NLV] = {0, 16700, 20900, 21950};

    // ---- Phase 2: bilinear scatter-accumulate (ds_add_f32) ------------------
    for (int i = tid; i < SAMPLES_PER_BLK; i += nthr) {
        const int m   = b * ROWS_PER_BLK + (i >> 4);
        const int rem = i & 15;
        const int lv  = rem >> 2;
        const int p   = rem & 3;
        const int l   = m / (NH * NQ);
        const int hq  = m - l * (NH * NQ);
        const int h   = hq / NQ;
        const int q   = hq - h * NQ;

        const long base = ((((long)l * NQ + q) * NH + h) * NLV + lv) * NP + p;
        const float x = sloc[base * 2 + 0];
        const float y = sloc[base * 2 + 1];
        const float w = awgt[base];

        const int   W   = Wl[lv];
        const int   H   = Hl[lv];
        const int   off = LSI[lv];
        const float cf  = x * (float)W;
        const float rf  = y * (float)H;
        const float clf = floorf(cf);
        const float rlf = floorf(rf);
        const float fc  = cf - clf;
        const float fr  = rf - rlf;
        const int   c0  = (int)clf;
        const int   r0  = (int)rlf;

        const float m00 = (fc - 1.0f) * (fr - 1.0f);  // (c0  , r0  )
        const float m01 = -(fc - 1.0f) * fr;          // (c0  , r0+1)
        const float m10 = -fc * (fr - 1.0f);          // (c0+1, r0  )
        const float m11 = fc * fr;                    // (c0+1, r0+1)

        #define MSDA_CORNER(cc, rr, mm)                                   \
            do {                                                          \
                const int c_ = (cc), r_ = (rr);                           \
                if (c_ >= 0 && c_ < W && r_ >= 0 && r_ < H)               \
                    atomicAdd(&smem[r_ * W + c_ + off], w * (mm));        \
            } while (0)

        MSDA_CORNER(c0,     r0,     m00);
        MSDA_CORNER(c0,     r0 + 1, m01);
        MSDA_CORNER(c0 + 1, r0,     m10);
        MSDA_CORNER(c0 + 1, r0 + 1, m11);
        #undef MSDA_CORNER
    }
    __syncthreads();

    // ---- Phase 3: bulk write-out -------------------------------------------
    if (contiguous) {
#if HAVE_TDM
        // One TDM descriptor per block: 1D tile of S_TOT dwords, LDS -> global.
        if (tid < 32) {   // single wave issues (EXEC ignored by TDM)
            const uint64_t ga = (uint64_t)(uintptr_t)(dst + (size_t)b * S_TOT);
            const uint32_t lo = (uint32_t)(uintptr_t)smem;     // low 32b = LDS offset
            u32x4 g0 = tdm_group0(lo, ga);
            i32x8 g1;
            g1[0] = 0x20000;                                   // data_size=2 (4B)
            g1[1] = (int)(((uint32_t)S_TOT & 0xFFFFu) << 16);  // tensor_dim0[15:0]
            g1[2] = (int)(((uint32_t)S_TOT >> 16) | (1u << 16)); // dim0[31:16] | tensor_dim1=1
            g1[3] = (int)((uint32_t)S_TOT << 16);              // tile_dim0 = S_TOT
            g1[4] = 0;                                         // tile_dim1/2 unused
            g1[5] = S_TOT;                                     // tensor_dim0_stride (unused, 1 row)
            g1[6] = 0;
            g1[7] = 0;
            const i32x4 z4 = {0, 0, 0, 0};
            const i32x8 z8 = {0, 0, 0, 0, 0, 0, 0, 0};
            __builtin_amdgcn_tensor_store_from_lds(g0, g1, z4, z4, z8, 0);
            __builtin_amdgcn_s_wait_tensorcnt(0);
        }
#else
        // Fallback: async LDS->global stores, contiguous across lanes.
        float* rb = dst + (size_t)b * S_TOT;
        for (int s = tid; s < S_TOT; s += nthr) {
            const uint64_t gaddr = (uint64_t)(uintptr_t)(rb + s);
            const uint32_t laddr = (uint32_t)(uintptr_t)(&smem[s]);
            asm volatile("global_store_async_from_lds_b32 %0, %1, off"
                         :: "v"(gaddr), "v"(laddr) : "memory");
        }
        asm volatile("s_wait_asynccnt 0x0" ::: "memory");
#endif
    } else {
        // Direct strided write to final (S, Q) layout (small-workspace fallback).
        float* colbase = dst + b;
        for (int s = tid; s < S_TOT; s += nthr) {
            const uint64_t gaddr = (uint64_t)(uintptr_t)(colbase + (size_t)s * NQ);
            const uint32_t laddr = (uint32_t)(uintptr_t)(&smem[s]);
            asm volatile("global_store_async_from_lds_b32 %0, %1, off"
                         :: "v"(gaddr), "v"(laddr) : "memory");
        }
        asm volatile("s_wait_asynccnt 0x0" ::: "memory");
    }
}

// =============================================================================
// Kernel 2: tiled transpose (300, S) -> (S, 300). Each block handles TS
// s-values x all 300 b-rows. TDM loads the strided 2D tile into LDS with
// hardware padding (pitch 33 dwords) so the transposed reads are conflict-free.
// =============================================================================
__global__ void msda_transpose_kernel(const float* __restrict__ ws,
                                      float* __restrict__ out) {
    __shared__ float tile[TROWS * TPITCH];   // 300*33*4 = 39600 B
    const int s0  = blockIdx.x * TS;
    const int tid = threadIdx.x;

#if HAVE_TDM
    if (tid < 32) {   // single wave issues the TDM load
        int xrem = S_TOT - s0;
        if (xrem > TS) xrem = TS;            // clamp tensor_dim0: OOB reads -> 0
        const uint64_t ga = (uint64_t)(uintptr_t)(ws + s0);
        const uint32_t lo = (uint32_t)(uintptr_t)tile;
        u32x4 g0 = tdm_group0(lo, ga);
        i32x8 g1;
        // data_size=4B | pad_enable | pad_interval=4 (32 dwords) | pad_amount=0 (1 dword)
        g1[0] = 0x20000 | (1 << 20) | (4 << 22);
        g1[1] = (int)(((uint32_t)xrem & 0xFFFFu) << 16);           // tensor_dim0[15:0]
        g1[2] = (int)((((uint32_t)xrem >> 16) & 0xFFFFu)
                      | (((uint32_t)TROWS & 0xFFFFu) << 16));      // dim0[31:16] | tensor_dim1
        g1[3] = (int)((uint32_t)TS << 16);                         // dim1[31:16]=0 | tile_dim0=32
        g1[4] = TROWS;                                             // tile_dim1=300, tile_dim2=0
        g1[5] = S_TOT;                                             // tensor_dim0_stride
        g1[6] = 0;
        g1[7] = 0;
        const i32x4 z4 = {0, 0, 0, 0};
        const i32x8 z8 = {0, 0, 0, 0, 0, 0, 0, 0};
        __builtin_amdgcn_tensor_load_to_lds(g0, g1, z4, z4, z8, 0);
        __builtin_amdgcn_s_wait_tensorcnt(0);
    }
    __syncthreads();
#else
    for (int idx = tid; idx < TROWS * TS; idx += blockDim.x) {
        const int bb = idx / TS;
        const int s  = idx - bb * TS;        // consecutive lanes -> consecutive s (coalesced)
        const int sg = s0 + s;
        tile[bb * TPITCH + s] = (sg < S_TOT) ? ws[(size_t)bb * S_TOT + sg] : 0.0f;
    }
    __syncthreads();
#endif

    // Transposed write-out: consecutive lanes -> consecutive b (coalesced).
    // LDS read stride = TPITCH = 33 (odd) -> conflict-free across 64 banks.
    for (int idx = tid; idx < TS * TROWS; idx += blockDim.x) {
        const int s  = idx / TROWS;
        const int bb = idx - s * TROWS;
        const int sg = s0 + s;
        if (sg < S_TOT)
            out[(size_t)sg * NQ + bb] = tile[bb * TPITCH + s];
    }
}

// =============================================================================
extern "C" void kernel_launch(void* const* d_in, const int* in_sizes, int n_in,
                              void* d_out, int out_size, void* d_ws, size_t ws_size,
                              hipStream_t stream) {
    const float* sloc = (const float*)d_in[0];  // (1,6,300,8,4,4,2) f32
    const float* awgt = (const float*)d_in[1];  // (1,6,300,8,4,4)   f32
    float* out = (float*)d_out;                 // (1,22223,300)     f32

    const size_t shmem   = (size_t)S_TOT * sizeof(float);          // 88892 B
    const size_t ws_need = (size_t)NQ * S_TOT * sizeof(float);     // 26.7 MB

    (void)hipFuncSetAttribute((const void*)msda_scatter_kernel,
                              hipFuncAttributeMaxDynamicSharedMemorySize,
                              (int)shmem);

    if (d_ws != nullptr && ws_size >= ws_need) {
        float* ws = (float*)d_ws;
        // Phase A: scatter + coalesced TDM store of each column to workspace.
        msda_scatter_kernel<<<dim3(NQ), dim3(256), shmem, stream>>>(sloc, awgt, ws, 1);
        // Phase B: tiled transpose to final (S, Q) layout, fully coalesced.
        dim3 g2((S_TOT + TS - 1) / TS);
        msda_transpose_kernel<<<g2, dim3(256), 0, stream>>>(ws, out);
    } else {
        // Workspace too small: direct strided write-out (still correct).
        msda_scatter_kernel<<<dim3(NQ), dim3(256), shmem, stream>>>(sloc, awgt, out, 0);
    }
}